// GraphTransformerLayer_38989713113821
// MI455X (gfx1250) — compile-verified
//
#include <hip/hip_runtime.h>
#include <math.h>

#define D     128
#define H     8
#define DFF   256
#define ROWS  80          // 100000 = 80 * 1250 ; 5 row-tiles of 16 per block
#define RT    5           // row tiles per block

typedef __attribute__((ext_vector_type(2))) float v2f;
typedef __attribute__((ext_vector_type(8))) float v8f;

// D(16x16,f32) += A(16x4,f32) * B(4x16,f32) ; full fp32 matrix pipe
__device__ __forceinline__ v8f wmma4(v2f a, v2f b, v8f c) {
    return __builtin_amdgcn_wmma_f32_16x16x4_f32(false, a, false, b, (short)0, c,
                                                 false, false);
}

// ---------------------------------------------------------------- zero fill
__global__ void gt_zero_kernel(float* __restrict__ p, int n) {
    int i = blockIdx.x * blockDim.x + threadIdx.x;
    if (i < n) p[i] = 0.0f;
}

// ---------------------------------------------------------------- QKV proj
// 80-row block: B fragment loaded once per k-step, reused by 5 row-tile WMMAs.
__global__ __launch_bounds__(256) void gt_qkv_kernel(
        const float* __restrict__ h,
        const float* __restrict__ Wq, const float* __restrict__ Wk,
        const float* __restrict__ Wv,
        float* __restrict__ Q, float* __restrict__ K, float* __restrict__ V) {
    __shared__ float sA[ROWS][132];               // rows 16B aligned (132*4=528)
    const int row0 = blockIdx.x * ROWS;
    const int tid  = threadIdx.x;
    {   // stage 80x128 h tile (2560 float4, 10 per thread)
        const float4* srcp = (const float4*)(h + (size_t)row0 * D);
        for (int i = tid; i < ROWS * 32; i += 256) {
            const int r = i >> 5, c4 = i & 31;
            const float4 v = srcp[i];
            sA[r][c4 * 4 + 0] = v.x; sA[r][c4 * 4 + 1] = v.y;
            sA[r][c4 * 4 + 2] = v.z; sA[r][c4 * 4 + 3] = v.w;
        }
    }
    __syncthreads();
    const int lane = tid & 31, wave = tid >> 5;
    const int half = lane >> 4, r = lane & 15;
    const int col0 = wave * 16;
    const float* Ws[3] = {Wq, Wk, Wv};
    float*       Os[3] = {Q, K, V};
#pragma unroll
    for (int m = 0; m < 3; ++m) {
        const float* __restrict__ W = Ws[m];
        v8f acc[RT];
#pragma unroll
        for (int t = 0; t < RT; ++t) acc[t] = (v8f){};
#pragma unroll
        for (int k = 0; k < D; k += 4) {
            const int ka = k + 2 * half;
            v2f b;
            b[0] = W[ka * D + col0 + r];
            b[1] = W[(ka + 1) * D + col0 + r];
#pragma unroll
            for (int t = 0; t < RT; ++t) {
                v2f a;
                a[0] = sA[16 * t + r][ka];
                a[1] = sA[16 * t + r][ka + 1];
                acc[t] = wmma4(a, b, acc[t]);
            }
        }
        float* O = Os[m] + (size_t)row0 * D;
#pragma unroll
        for (int t = 0; t < RT; ++t)
#pragma unroll
            for (int i = 0; i < 8; ++i)
                O[(size_t)(16 * t + i + 8 * half) * D + col0 + r] = acc[t][i];
    }
}

// --------------------------------------------------- edge attention+scatter
__global__ __launch_bounds__(256) void gt_edge_kernel(
        const float* __restrict__ Q, const float* __restrict__ K,
        const float* __restrict__ V,
        const int* __restrict__ src, const int* __restrict__ dst,
        float* __restrict__ wV, float* __restrict__ z, int E) {
    const int lane = threadIdx.x & 31;
    const int e = blockIdx.x * 8 + (threadIdx.x >> 5);   // one wave per edge
    if (e >= E) return;
    const int s = src[e], d = dst[e];
    const float4 kv = ((const float4*)(K + (size_t)s * D))[lane];
    const float4 qv = ((const float4*)(Q + (size_t)d * D))[lane];
    float p = kv.x * qv.x + kv.y * qv.y + kv.z * qv.z + kv.w * qv.w;
    // quad reduce: lanes 4h..4h+3 form head h (DH=16 = 4 lanes * 4 floats)
    p += __shfl_xor(p, 1);
    p += __shfl_xor(p, 2);
    const float sc = __expf(fminf(fmaxf(p * 0.25f, -5.0f), 5.0f)); // 1/sqrt(16)
    const float4 vv = ((const float4*)(V + (size_t)s * D))[lane];
    float* wdst = wV + (size_t)d * D + lane * 4;
    atomicAdd(wdst + 0, vv.x * sc);
    atomicAdd(wdst + 1, vv.y * sc);
    atomicAdd(wdst + 2, vv.z * sc);
    atomicAdd(wdst + 3, vv.w * sc);
    if ((lane & 3) == 0) atomicAdd(z + (size_t)d * H + (lane >> 2), sc);
}

// -------------------------------- shared LayerNorm epilogue over sY -> dst
__device__ __forceinline__ void layernorm80(
        float (*sY)[132], float (*sRed)[ROWS][2], float* sMu, float* sRs,
        const float* __restrict__ g, const float* __restrict__ b,
        float* __restrict__ dstp, int row0, int tid) {
    if (tid < 2 * ROWS) {                 // 2 threads per row, 64 cols each
        const int rr = tid >> 1, part = tid & 1;
        float s = 0.f, ss = 0.f;
#pragma unroll 8
        for (int j = 0; j < 64; ++j) {
            const float v = sY[rr][part * 64 + j];
            s += v; ss += v * v;
        }
        sRed[0][rr][part] = s; sRed[1][rr][part] = ss;
    }
    __syncthreads();
    if (tid < ROWS) {
        const float s  = sRed[0][tid][0] + sRed[0][tid][1];
        const float ss = sRed[1][tid][0] + sRed[1][tid][1];
        const float mu  = s * (1.0f / 128.0f);
        const float var = ss * (1.0f / 128.0f) - mu * mu;
        sMu[tid] = mu;
        sRs[tid] = rsqrtf(var + 1e-5f);
    }
    __syncthreads();
    for (int i = tid; i < ROWS * 128; i += 256) {
        const int rr = i >> 7, cc = i & 127;
        dstp[(size_t)(row0 + rr) * D + cc] =
            (sY[rr][cc] - sMu[rr]) * sRs[rr] * g[cc] + b[cc];
    }
}

// ---------------------------------------- output proj + residual + LN1 -> x
__global__ __launch_bounds__(256) void gt_attnout_kernel(
        const float* __restrict__ hin,
        const float* __restrict__ wV, const float* __restrict__ z,
        const float* __restrict__ Wo, const float* __restrict__ bo,
        const float* __restrict__ g1, const float* __restrict__ bb1,
        float* __restrict__ x) {
    __shared__ float sA[ROWS][132];
    __shared__ float sY[ROWS][132];
    __shared__ float sRed[2][ROWS][2];
    __shared__ float sMu[ROWS], sRs[ROWS];
    const int row0 = blockIdx.x * ROWS;
    const int tid  = threadIdx.x;
    for (int i = tid; i < ROWS * 128; i += 256) {      // h_attn = wV/(z+eps)
        const int r = i >> 7, c = i & 127;
        const float zz = z[(size_t)(row0 + r) * H + (c >> 4)];
        sA[r][c] = wV[(size_t)(row0 + r) * D + c] / (zz + 1e-6f);
    }
    __syncthreads();
    const int lane = tid & 31, wave = tid >> 5;
    const int half = lane >> 4, r = lane & 15;
    const int col0 = wave * 16;
    v8f acc[RT];
#pragma unroll
    for (int t = 0; t < RT; ++t) acc[t] = (v8f){};
#pragma unroll
    for (int k = 0; k < D; k += 4) {
        const int ka = k + 2 * half;
        v2f b;
        b[0] = Wo[ka * D + col0 + r];
        b[1] = Wo[(ka + 1) * D + col0 + r];
#pragma unroll
        for (int t = 0; t < RT; ++t) {
            v2f a;
            a[0] = sA[16 * t + r][ka];
            a[1] = sA[16 * t + r][ka + 1];
            acc[t] = wmma4(a, b, acc[t]);
        }
    }
#pragma unroll
    for (int t = 0; t < RT; ++t)
#pragma unroll
        for (int i = 0; i < 8; ++i) {
            const int rr = 16 * t + i + 8 * half, cc = col0 + r;
            sY[rr][cc] = hin[(size_t)(row0 + rr) * D + cc] + acc[t][i] + bo[cc];
        }
    __syncthreads();
    layernorm80(sY, sRed, sMu, sRs, g1, bb1, x, row0, tid);
}

// ----------------------------------------- fused FFN + residual + LN2 -> out
__global__ __launch_bounds__(256) void gt_ffn_kernel(
        const float* __restrict__ x,
        const float* __restrict__ W1, const float* __restrict__ b1,
        const float* __restrict__ W2, const float* __restrict__ b2,
        const float* __restrict__ g2, const float* __restrict__ bb2,
        float* __restrict__ out) {
    __shared__ float sX[ROWS][132];
    __shared__ float sT[ROWS][264];
    __shared__ float sY[ROWS][132];
    __shared__ float sRed[2][ROWS][2];
    __shared__ float sMu[ROWS], sRs[ROWS];
    const int row0 = blockIdx.x * ROWS;
    const int tid  = threadIdx.x;
    {
        const float4* srcp = (const float4*)(x + (size_t)row0 * D);
        for (int i = tid; i < ROWS * 32; i += 256) {
            const int r = i >> 5, c4 = i & 31;
            const float4 v = srcp[i];
            sX[r][c4 * 4 + 0] = v.x; sX[r][c4 * 4 + 1] = v.y;
            sX[r][c4 * 4 + 2] = v.z; sX[r][c4 * 4 + 3] = v.w;
        }
    }
    __syncthreads();
    const int lane = tid & 31, wave = tid >> 5;
    const int half = lane >> 4, r = lane & 15;
    {   // GEMM1: x @ W1 (+b1, relu); wave covers 256-col tiles w and w+8
        const int c0a = wave * 16, c0b = (wave + 8) * 16;
        v8f acc0[RT], acc1[RT];
#pragma unroll
        for (int t = 0; t < RT; ++t) { acc0[t] = (v8f){}; acc1[t] = (v8f){}; }
#pragma unroll
        for (int k = 0; k < D; k += 4) {
            const int ka = k + 2 * half;
            v2f ba, bb;
            ba[0] = W1[ka * DFF + c0a + r]; ba[1] = W1[(ka + 1) * DFF + c0a + r];
            bb[0] = W1[ka * DFF + c0b + r]; bb[1] = W1[(ka + 1) * DFF + c0b + r];
#pragma unroll
            for (int t = 0; t < RT; ++t) {
                v2f a;
                a[0] = sX[16 * t + r][ka];
                a[1] = sX[16 * t + r][ka + 1];
                acc0[t] = wmma4(a, ba, acc0[t]);
                acc1[t] = wmma4(a, bb, acc1[t]);
            }
        }
#pragma unroll
        for (int t = 0; t < RT; ++t)
#pragma unroll
            for (int i = 0; i < 8; ++i) {
                const int rr = 16 * t + i + 8 * half;
                sT[rr][c0a + r] = fmaxf(acc0[t][i] + b1[c0a + r], 0.0f);
                sT[rr][c0b + r] = fmaxf(acc1[t][i] + b1[c0b + r], 0.0f);
            }
    }
    __syncthreads();
    {   // GEMM2: t @ W2 (+b2, +x residual)
        const int col0 = wave * 16;
        v8f acc[RT];
#pragma unroll
        for (int t = 0; t < RT; ++t) acc[t] = (v8f){};
#pragma unroll
        for (int k = 0; k < DFF; k += 4) {
            const int ka = k + 2 * half;
            v2f b;
            b[0] = W2[ka * D + col0 + r];
            b[1] = W2[(ka + 1) * D + col0 + r];
#pragma unroll
            for (int t = 0; t < RT; ++t) {
                v2f a;
                a[0] = sT[16 * t + r][ka];
                a[1] = sT[16 * t + r][ka + 1];
                acc[t] = wmma4(a, b, acc[t]);
            }
        }
#pragma unroll
        for (int t = 0; t < RT; ++t)
#pragma unroll
            for (int i = 0; i < 8; ++i) {
                const int rr = 16 * t + i + 8 * half, cc = col0 + r;
                sY[rr][cc] = sX[rr][cc] + acc[t][i] + b2[cc];
            }
    }
    __syncthreads();
    layernorm80(sY, sRed, sMu, sRs, g2, bb2, out, row0, tid);
}

// ---------------------------------------------------------------- launcher
extern "C" void kernel_launch(void* const* d_in, const int* in_sizes, int n_in,
                              void* d_out, int out_size, void* d_ws, size_t ws_size,
                              hipStream_t stream) {
    const float* h    = (const float*)d_in[0];
    const int*   src  = (const int*)  d_in[1];
    const int*   dst  = (const int*)  d_in[2];
    const float* Wq   = (const float*)d_in[3];
    const float* Wk   = (const float*)d_in[4];
    const float* Wv   = (const float*)d_in[5];
    const float* Wo   = (const float*)d_in[6];
    const float* bo   = (const float*)d_in[7];
    const float* ln1g = (const float*)d_in[8];
    const float* ln1b = (const float*)d_in[9];
    const float* W1   = (const float*)d_in[10];
    const float* b1   = (const float*)d_in[11];
    const float* W2   = (const float*)d_in[12];
    const float* b2   = (const float*)d_in[13];
    const float* ln2g = (const float*)d_in[14];
    const float* ln2b = (const float*)d_in[15];

    const int N = in_sizes[0] / D;     // 100000 = 80 * 1250
    const int E = in_sizes[1];         // 1600000

    float* Q  = (float*)d_ws;
    float* K  = Q  + (size_t)N * D;
    float* V  = K  + (size_t)N * D;
    float* wV = V  + (size_t)N * D;
    float* z  = wV + (size_t)N * D;    // contiguous with wV
    float* x  = z  + (size_t)N * H;

    const int zn = N * (D + H);        // wV + z together
    gt_zero_kernel<<<(zn + 255) / 256, 256, 0, stream>>>(wV, zn);
    gt_qkv_kernel<<<N / ROWS, 256, 0, stream>>>(h, Wq, Wk, Wv, Q, K, V);
    gt_edge_kernel<<<(E + 7) / 8, 256, 0, stream>>>(Q, K, V, src, dst, wV, z, E);
    gt_attnout_kernel<<<N / ROWS, 256, 0, stream>>>(h, wV, z, Wo, bo, ln1g, ln1b, x);
    gt_ffn_kernel<<<N / ROWS, 256, 0, stream>>>(x, W1, b1, W2, b2, ln2g, ln2b,
                                                (float*)d_out);
}